// QuantumConv2d_87771951661102
// MI455X (gfx1250) — compile-verified
//
#include <hip/hip_runtime.h>
#include <math.h>

// Problem geometry (fixed by the reference)
#define HW_IN   (512 * 512)   // per-channel pixels
#define HW_OUT  (256 * 256)   // per-channel output pixels / positions per batch
#define PI_F    3.14159265358979f
#define EPS_F   1e-8f

typedef float v2f __attribute__((ext_vector_type(2)));
typedef float v8f __attribute__((ext_vector_type(8)));

// ---- fast transcendentals on the TRANS pipe ----
__device__ __forceinline__ float fast_tanh(float x) {
#if __has_builtin(__builtin_amdgcn_tanhf)
    return __builtin_amdgcn_tanhf(x);            // v_tanh_f32 (gfx1250 TRANS op)
#else
    // tanh(x) = 1 - 2/(exp(2x)+1); |x| <= ~2.3 here, no overflow issues
    float e = __expf(2.0f * x);                  // v_exp_f32 path
    return 1.0f - 2.0f * __builtin_amdgcn_rcpf(e + 1.0f);
#endif
}

// ---- order-preserving float<->uint mapping (for integer min/max atomics) ----
__device__ __forceinline__ unsigned f2ord(float f) {
    unsigned u = __float_as_uint(f);
    return (u & 0x80000000u) ? ~u : (u | 0x80000000u);
}
__device__ __forceinline__ float ord2f(unsigned u) {
    return __uint_as_float((u & 0x80000000u) ? (u ^ 0x80000000u) : ~u);
}

// ws layout: [0]=min (ordered uint), [1]=max (ordered uint),
//            floats [2..25] = per-(layer,qubit) {cos(w0), sin(w1), w2}
__global__ void qc_init_kernel(const float* __restrict__ weights,
                               unsigned* __restrict__ ws_u,
                               float* __restrict__ ws_f) {
    int t = threadIdx.x;
    if (t == 0) { ws_u[0] = 0xFFFFFFFFu; ws_u[1] = 0u; }
    if (t < 8) {                    // 2 layers x 4 qubits (precise libm, runs once)
        int base = t * 3;
        ws_f[2 + base + 0] = cosf(weights[base + 0]);
        ws_f[2 + base + 1] = sinf(weights[base + 1]);
        ws_f[2 + base + 2] = weights[base + 2];
    }
}

// ---- pass 1: global min/max of grayscale (== min/max of all patch values) ----
__global__ void __launch_bounds__(256)
qc_minmax_kernel(const float* __restrict__ x, unsigned* __restrict__ ws_u) {
    int t  = blockIdx.x * 256 + threadIdx.x;
    int i0 = t * 4;                       // pixel index into (B, H*W), 4 px/thread
    int b  = i0 >> 18;                    // / 262144
    int r  = i0 & (HW_IN - 1);
    const float4 a = *(const float4*)(x + (b * 3 + 0) * HW_IN + r);
    const float4 c = *(const float4*)(x + (b * 3 + 1) * HW_IN + r);
    const float4 d = *(const float4*)(x + (b * 3 + 2) * HW_IN + r);
    const float k3 = 1.0f / 3.0f;
    float g0 = (a.x + c.x + d.x) * k3;
    float g1 = (a.y + c.y + d.y) * k3;
    float g2 = (a.z + c.z + d.z) * k3;
    float g3 = (a.w + c.w + d.w) * k3;
    float mn = fminf(fminf(g0, g1), fminf(g2, g3));
    float mx = fmaxf(fmaxf(g0, g1), fmaxf(g2, g3));

    // wave32 reduction
    #pragma unroll
    for (int off = 16; off >= 1; off >>= 1) {
        mn = fminf(mn, __shfl_xor(mn, off));
        mx = fmaxf(mx, __shfl_xor(mx, off));
    }
    __shared__ float smn[8], smx[8];
    int lane = threadIdx.x & 31, wid = threadIdx.x >> 5;
    if (lane == 0) { smn[wid] = mn; smx[wid] = mx; }
    __syncthreads();
    if (threadIdx.x == 0) {
        float bmn = smn[0], bmx = smx[0];
        #pragma unroll
        for (int i = 1; i < 8; ++i) { bmn = fminf(bmn, smn[i]); bmx = fmaxf(bmx, smx[i]); }
        atomicMin(&ws_u[0], f2ord(bmn));
        atomicMax(&ws_u[1], f2ord(bmx));
    }
}

// ---- pass 2: fused grayscale + normalize + circuit + WMMA projection ----
__global__ void __launch_bounds__(256)
qc_main_kernel(const float* __restrict__ x,
               const float* __restrict__ proj_w,   // (4,Q,1,1) -> [o*4+q]
               const float* __restrict__ proj_b,   // (4,)
               const unsigned* __restrict__ ws_u,
               const float* __restrict__ ws_f,
               float* __restrict__ out) {          // (B,4,256,256)
    const int lane = threadIdx.x & 31;
    const int p    = blockIdx.x * 256 + threadIdx.x;   // position id in [0, B*HW_OUT)
    const int b    = p >> 16;
    const int hw   = p & (HW_OUT - 1);
    const int h    = hw >> 8;
    const int w    = hw & 255;

    // 2x2 patch over 3 channels -> grayscale (coalesced float2 loads)
    const float* xb = x + b * 3 * HW_IN;
    const int row0 = (2 * h) * 512 + 2 * w;
    const int row1 = row0 + 512;
    float2 c0r0 = *(const float2*)(xb + 0 * HW_IN + row0);
    float2 c0r1 = *(const float2*)(xb + 0 * HW_IN + row1);
    float2 c1r0 = *(const float2*)(xb + 1 * HW_IN + row0);
    float2 c1r1 = *(const float2*)(xb + 1 * HW_IN + row1);
    float2 c2r0 = *(const float2*)(xb + 2 * HW_IN + row0);
    float2 c2r1 = *(const float2*)(xb + 2 * HW_IN + row1);
    const float k3 = 1.0f / 3.0f;
    float g0 = (c0r0.x + c1r0.x + c2r0.x) * k3;   // q0 = (r0,c0)
    float g1 = (c0r0.y + c1r0.y + c2r0.y) * k3;   // q1 = (r0,c1)
    float g2 = (c0r1.x + c1r1.x + c2r1.x) * k3;   // q2 = (r1,c0)
    float g3 = (c0r1.y + c1r1.y + c2r1.y) * k3;   // q3 = (r1,c1)

    const float pmin  = ord2f(ws_u[0]);
    const float pmax  = ord2f(ws_u[1]);
    const float inv   = 1.0f / (pmax - pmin + EPS_F);
    const float scale = inv * PI_F;               // sin( (g-pmin)*inv * PI )
    const float bias  = -pmin * scale;

    // hardware v_sin_f32 via __sinf (arg in [0, PI], well inside accurate range)
    float e0 = __sinf(fmaf(g0, scale, bias));
    float e1 = __sinf(fmaf(g1, scale, bias));
    float e2 = __sinf(fmaf(g2, scale, bias));
    float e3 = __sinf(fmaf(g3, scale, bias));

    // two mixing layers: m[q] = e[q]*cos(w0) + e[(q+1)%4]*sin(w1) + w2 ; e = tanh(m)
    #pragma unroll
    for (int l = 0; l < 2; ++l) {
        const float* tw = ws_f + 2 + l * 12;
        float m0 = fmaf(e0, tw[0], fmaf(e1, tw[1],  tw[2]));
        float m1 = fmaf(e1, tw[3], fmaf(e2, tw[4],  tw[5]));
        float m2 = fmaf(e2, tw[6], fmaf(e3, tw[7],  tw[8]));
        float m3 = fmaf(e3, tw[9], fmaf(e0, tw[10], tw[11]));
        e0 = fast_tanh(m0); e1 = fast_tanh(m1);
        e2 = fast_tanh(m2); e3 = fast_tanh(m3);
    }

    // ---------- projection via V_WMMA_F32_16X16X4_F32 ----------
    // D(16x16) = A(16x4: 16 positions x 4 qubits) * B(4x16: qubit x out-ch) + C(bias)
    // A layout: lanes 0-15 hold (M=lane, K=0/1) in v0/v1; lanes 16-31 hold (M=lane-16, K=2/3).
    // B layout: v0: lanes 0-15 K=0, lanes 16-31 K=2; v1: K=1 / K=3; N = lane&15.
    // C/D layout: VGPR r: lanes 0-15 -> (M=r, N=lane); lanes 16-31 -> (M=r+8, N=lane-16).
    const int  nlo = lane & 15;
    const bool hi  = lane >= 16;

    v2f Bm;
    float cbias;
    if (nlo < 4) {                                  // N = output channel o
        Bm[0] = proj_w[nlo * 4 + (hi ? 2 : 0)];     // B[k][n] = proj_w[o][q]
        Bm[1] = proj_w[nlo * 4 + (hi ? 3 : 1)];
        cbias = proj_b[nlo];
    } else {
        Bm[0] = 0.0f; Bm[1] = 0.0f; cbias = 0.0f;   // dead columns
    }
    v8f Cm;
    #pragma unroll
    for (int r = 0; r < 8; ++r) Cm[r] = cbias;

    // Repack this wave's 32 positions (one per lane) into two A tiles of 16 rows.
    float s2 = __shfl(e2, nlo);        // enc2 of lane (lane&15)
    float s3 = __shfl(e3, nlo);
    v2f A0; A0[0] = hi ? s2 : e0;  A0[1] = hi ? s3 : e1;   // rows = lanes 0..15
    float s0 = __shfl(e0, nlo + 16);   // enc0 of lane (lane&15)+16
    float s1 = __shfl(e1, nlo + 16);
    v2f A1; A1[0] = hi ? e2 : s0;  A1[1] = hi ? e3 : s1;   // rows = lanes 16..31

    v8f D0 = __builtin_amdgcn_wmma_f32_16x16x4_f32(false, A0, false, Bm, (short)0, Cm, false, false);
    v8f D1 = __builtin_amdgcn_wmma_f32_16x16x4_f32(false, A1, false, Bm, (short)0, Cm, false, false);

    // store: active lanes are N<4; each holds 8 consecutive positions of one channel
    const int p0 = blockIdx.x * 256 + (threadIdx.x & ~31);   // wave's first position
    if (nlo < 4) {
        const int o = nlo;
        {   // group 0: positions p0 .. p0+15
            int hwb = ((p0) & (HW_OUT - 1)) + (hi ? 8 : 0);
            float* op = out + (b * 4 + o) * HW_OUT + hwb;
            *(float4*)(op + 0) = make_float4(D0[0], D0[1], D0[2], D0[3]);
            *(float4*)(op + 4) = make_float4(D0[4], D0[5], D0[6], D0[7]);
        }
        {   // group 1: positions p0+16 .. p0+31
            int hwb = ((p0 + 16) & (HW_OUT - 1)) + (hi ? 8 : 0);
            float* op = out + (b * 4 + o) * HW_OUT + hwb;
            *(float4*)(op + 0) = make_float4(D1[0], D1[1], D1[2], D1[3]);
            *(float4*)(op + 4) = make_float4(D1[4], D1[5], D1[6], D1[7]);
        }
    }
}

extern "C" void kernel_launch(void* const* d_in, const int* in_sizes, int n_in,
                              void* d_out, int out_size, void* d_ws, size_t ws_size,
                              hipStream_t stream) {
    const float* x       = (const float*)d_in[0];   // (64,3,512,512)
    const float* weights = (const float*)d_in[1];   // (2,4,3)
    const float* proj_w  = (const float*)d_in[2];   // (4,4,1,1)
    const float* proj_b  = (const float*)d_in[3];   // (4,)
    float*    out  = (float*)d_out;                 // (64,4,256,256)
    unsigned* ws_u = (unsigned*)d_ws;
    float*    ws_f = (float*)d_ws;

    qc_init_kernel<<<1, 32, 0, stream>>>(weights, ws_u, ws_f);

    // 64*512*512 = 16,777,216 gray pixels, 4 per thread -> 4,194,304 threads
    qc_minmax_kernel<<<16384, 256, 0, stream>>>(x, ws_u);

    // 64*256*256 = 4,194,304 positions, 1 per thread
    qc_main_kernel<<<16384, 256, 0, stream>>>(x, proj_w, proj_b, ws_u, ws_f, out);
}